// StackedGCNResidual_71751723647605
// MI455X (gfx1250) — compile-verified
//
#include <hip/hip_runtime.h>
#include <hip/hip_bf16.h>

typedef __attribute__((ext_vector_type(2))) float v2f;
typedef __attribute__((ext_vector_type(8))) float v8f;

#define HID 64
#define OUTC 3

// ---------------- degree / norm ----------------
__global__ void deg_init_kernel(int* deg, int n) {
    int i = blockIdx.x * blockDim.x + threadIdx.x;
    if (i < n) deg[i] = 1;                       // self-loop contributes 1
}

__global__ void deg_count_kernel(const int* __restrict__ ei, int* deg, int nedges) {
    int e = blockIdx.x * blockDim.x + threadIdx.x;
    if (e < nedges) atomicAdd(&deg[ei[nedges + e]], 1);   // dst row
}

__global__ void dinv_kernel(float* dinv_io, int n) {
    int i = blockIdx.x * blockDim.x + threadIdx.x;
    if (i < n) {
        int d = ((const int*)dinv_io)[i];
        dinv_io[i] = rsqrtf((float)d);           // deg >= 1 always
    }
}

// ---------------- fp32 WMMA GEMM: H[N,64] = X[N,64] @ W[64,64] ----------------
// One wave computes a 16-row x 64-col tile via V_WMMA_F32_16X16X4_F32.
// A-matrix 16x4 layout: lane(0-15) row=M, VGPRs hold K={0,1}; lanes 16-31 K={2,3}.
// C/D 16x16 layout: VGPR r holds M=r (lanes 0-15) / M=8+r (lanes 16-31), N=lane&15.
__launch_bounds__(128)
__global__ void gemm64_wmma(const float* __restrict__ X,
                            const float* __restrict__ W,
                            float* __restrict__ H, int nrows) {
    __shared__ float wlds[HID * HID];            // 16 KB
    int t = threadIdx.x;
    for (int i = t * 4; i < HID * HID; i += 128 * 4)
        *(float4*)(wlds + i) = *(const float4*)(W + i);
    __syncthreads();

    int wave = t >> 5;
    int lane = t & 31;
    int tile = blockIdx.x * 4 + wave;            // 16-row tile index
    if (tile * 16 >= nrows) return;              // wave-uniform: EXEC stays all-1s

    int mrow  = tile * 16 + (lane & 15);
    int khalf = lane >> 4;                       // 0 or 1
    int col   = lane & 15;

    v8f acc0 = {}, acc1 = {}, acc2 = {}, acc3 = {};
    for (int k0 = 0; k0 < HID; k0 += 4) {
        int kk = k0 + khalf * 2;
        v2f a;
        a.x = X[mrow * HID + kk];
        a.y = X[mrow * HID + kk + 1];
        v2f b0, b1, b2, b3;
        b0.x = wlds[kk * HID + col];           b0.y = wlds[(kk + 1) * HID + col];
        b1.x = wlds[kk * HID + 16 + col];      b1.y = wlds[(kk + 1) * HID + 16 + col];
        b2.x = wlds[kk * HID + 32 + col];      b2.y = wlds[(kk + 1) * HID + 32 + col];
        b3.x = wlds[kk * HID + 48 + col];      b3.y = wlds[(kk + 1) * HID + 48 + col];
        acc0 = __builtin_amdgcn_wmma_f32_16x16x4_f32(false, a, false, b0, (short)0, acc0, false, false);
        acc1 = __builtin_amdgcn_wmma_f32_16x16x4_f32(false, a, false, b1, (short)0, acc1, false, false);
        acc2 = __builtin_amdgcn_wmma_f32_16x16x4_f32(false, a, false, b2, (short)0, acc2, false, false);
        acc3 = __builtin_amdgcn_wmma_f32_16x16x4_f32(false, a, false, b3, (short)0, acc3, false, false);
    }

    int rbase = tile * 16 + khalf * 8;
    for (int r = 0; r < 8; ++r) {
        H[(rbase + r) * HID +  0 + col] = acc0[r];
        H[(rbase + r) * HID + 16 + col] = acc1[r];
        H[(rbase + r) * HID + 32 + col] = acc2[r];
        H[(rbase + r) * HID + 48 + col] = acc3[r];
    }
}

// ---------------- AGG = b + h * dinv^2 (bias + self-loop) ----------------
__global__ void agg_init_kernel(const float* __restrict__ H,
                                const float* __restrict__ dinv,
                                const float* __restrict__ b,
                                float* __restrict__ AGG, int total) {
    int i = blockIdx.x * blockDim.x + threadIdx.x;
    if (i >= total) return;
    int node = i >> 6;
    int c    = i & 63;
    float w = dinv[node];
    AGG[i] = b[c] + H[i] * w * w;
}

// ---------------- edge scatter: AGG[dst] += H[src] * dinv[src]*dinv[dst] ----------------
__launch_bounds__(256)
__global__ void edge_scatter_kernel(const int* __restrict__ ei,
                                    const float* __restrict__ dinv,
                                    const float* __restrict__ H,
                                    float* __restrict__ AGG, int nedges) {
    int tid = blockIdx.x * blockDim.x + threadIdx.x;   // 16 threads per edge
    int e  = tid >> 4;
    if (e >= nedges) return;
    int c4 = (tid & 15) * 4;
    int s = ei[e];
    int d = ei[nedges + e];
    float w = dinv[s] * dinv[d];
    float4 hv = *(const float4*)(H + s * HID + c4);
    float* dst = AGG + d * HID + c4;
    atomicAdd(dst + 0, hv.x * w);
    atomicAdd(dst + 1, hv.y * w);
    atomicAdd(dst + 2, hv.z * w);
    atomicAdd(dst + 3, hv.w * w);
}

// ---------------- relu(+residual) ----------------
__global__ void post_kernel(const float* __restrict__ AGG,
                            const float* __restrict__ Xres,
                            float* __restrict__ Xout, int total, int use_res) {
    int i = blockIdx.x * blockDim.x + threadIdx.x;
    if (i >= total) return;
    float v = AGG[i] + (use_res ? Xres[i] : 0.0f);
    Xout[i] = v > 0.0f ? v : 0.0f;
}

// ---------------- classifier: out[N,3] = X @ Wc + bc ----------------
__global__ void classifier_kernel(const float* __restrict__ X,
                                  const float* __restrict__ Wc,
                                  const float* __restrict__ bc,
                                  float* __restrict__ out, int n) {
    int i = blockIdx.x * blockDim.x + threadIdx.x;
    if (i >= n) return;
    float a0 = bc[0], a1 = bc[1], a2 = bc[2];
    const float* xr = X + i * HID;
    for (int c = 0; c < HID; ++c) {
        float v = xr[c];
        a0 += v * Wc[c * OUTC + 0];
        a1 += v * Wc[c * OUTC + 1];
        a2 += v * Wc[c * OUTC + 2];
    }
    out[i * OUTC + 0] = a0;
    out[i * OUTC + 1] = a1;
    out[i * OUTC + 2] = a2;
}

extern "C" void kernel_launch(void* const* d_in, const int* in_sizes, int n_in,
                              void* d_out, int out_size, void* d_ws, size_t ws_size,
                              hipStream_t stream) {
    const float* x   = (const float*)d_in[0];     // [N,64]
    const int*   ei  = (const int*)d_in[1];       // [2,E]
    const float* Ws  = (const float*)d_in[2];     // [4,64,64]
    const float* bs  = (const float*)d_in[3];     // [4,64]
    const float* Wc  = (const float*)d_in[4];     // [64,3]
    const float* bc  = (const float*)d_in[5];     // [3]
    float* out = (float*)d_out;

    const int N = in_sizes[0] / HID;
    const int E = in_sizes[1] / 2;
    const int NF = N * HID;

    float* P0   = (float*)d_ws;
    float* P1   = P0 + NF;
    float* AGG  = P1 + NF;
    float* dinv = AGG + NF;                       // reused: int deg -> float dinv

    const int TB = 256;
    // degrees + symmetric norm factors
    deg_init_kernel<<<(N + TB - 1) / TB, TB, 0, stream>>>((int*)dinv, N);
    deg_count_kernel<<<(E + TB - 1) / TB, TB, 0, stream>>>(ei, (int*)dinv, E);
    dinv_kernel<<<(N + TB - 1) / TB, TB, 0, stream>>>(dinv, N);

    const int ntiles = (N + 15) / 16;             // 6250 (exact)
    const int gemm_blocks = (ntiles + 3) / 4;
    const int nf_blocks   = (NF + TB - 1) / TB;
    const int es_threads  = E * 16;
    const int es_blocks   = (es_threads + TB - 1) / TB;

    const float* xin = x;
    for (int L = 0; L < 4; ++L) {
        float* h = (xin == (const float*)P0) ? P1 : P0;   // buffer not holding xin
        gemm64_wmma<<<gemm_blocks, 128, 0, stream>>>(xin, Ws + L * HID * HID, h, N);
        agg_init_kernel<<<nf_blocks, TB, 0, stream>>>(h, dinv, bs + L * HID, AGG, NF);
        edge_scatter_kernel<<<es_blocks, TB, 0, stream>>>(ei, dinv, h, AGG, E);
        float* xnext = (L == 0) ? P1 : h;                 // L>0: overwrite dead h
        post_kernel<<<nf_blocks, TB, 0, stream>>>(AGG, xin, xnext, NF, L > 0 ? 1 : 0);
        xin = xnext;
    }

    classifier_kernel<<<(N + TB - 1) / TB, TB, 0, stream>>>(xin, Wc, bc, out, N);
}